// BKT_RNN_45002667327870
// MI455X (gfx1250) — compile-verified
//
#include <hip/hip_runtime.h>
#include <hip/hip_bf16.h>

typedef float v2f __attribute__((ext_vector_type(2)));
typedef float v8f __attribute__((ext_vector_type(8)));

constexpr int Tn = 1024;
constexpr int Bn = 4096;
constexpr int WAVES_PER_BLK = 8;         // 256 threads
constexpr int CHAINS_PER_WAVE = 16;      // one WMMA N-column per chain
constexpr int NBLOCKS = Bn / (WAVES_PER_BLK * CHAINS_PER_WAVE);  // 32

// Native V_TANH_F32 (gfx1250); branch-free fallback otherwise.
__device__ __forceinline__ float fast_tanh(float v) {
#if __has_builtin(__builtin_amdgcn_tanhf)
    return __builtin_amdgcn_tanhf(v);
#else
    const float e = __expf(2.0f * v);
    return 1.0f - 2.0f / (e + 1.0f);
#endif
}

// Raw v_log_f32 (log2) * ln2. Inputs here are provably normal (correct is
// bounded away from 0/1 since |preact| <= ~2 -> g,s in [0.02,0.98]), and the
// -100 clamp absorbs any edge case, so the OCML denorm pre-scale is dead code.
__device__ __forceinline__ float fast_log(float v) {
#if __has_builtin(__builtin_amdgcn_logf)
    return __builtin_amdgcn_logf(v) * 0.69314718055994531f;
#else
    return __logf(v);
#endif
}

// Fused RNN + BKT scan. One wave = 16 chains.
// Wave state: h kept in WMMA B-matrix layout (4x16 f32, 2 VGPRs):
//   B VGPR0: lanes 0-15 = h0[chain], lanes 16-31 = h2[chain]
//   B VGPR1: lanes 0-15 = h1[chain], lanes 16-31 = h3[chain]
// A = W_hh zero-padded to 16x4 (constant):
//   A VGPR0: lane m = W_hh[m][0] (lo) / W_hh[m][2] (hi), rows m>=4 zero
//   A VGPR1: lane m = W_hh[m][1] (lo) / W_hh[m][3] (hi)
// D = A*B: VGPR r, lanes 0-15 = D[r][chain]; rows 0-3 are the new
// pre-activations (chain per lane), re-striped into B layout with bpermutes.
__global__ __launch_bounds__(256) void bkt_rnn_fused(
    const float* __restrict__ x, const float* __restrict__ y,
    const float* __restrict__ prior, const float* __restrict__ W_ih,
    const float* __restrict__ W_hh, const float* __restrict__ b_ih,
    const float* __restrict__ b_hh,
    float* __restrict__ out_correct, float* __restrict__ out_latent,
    float* __restrict__ block_loss)
{
    const int lane      = threadIdx.x & 31;
    const int waveInBlk = threadIdx.x >> 5;
    const int n         = lane & 15;     // chain-in-wave (lo), mirrored in hi
    const int half      = lane >> 4;     // 0: K=0/1 rows, 1: K=2/3 rows
    const int chain     = (blockIdx.x * WAVES_PER_BLK + waveInBlk) * CHAINS_PER_WAVE + n;

    // ---- constant A matrix (W_hh padded 16x4), ISA A-layout ----
    v2f a;
    a.x = (n < 4) ? W_hh[n * 4 + (half ? 2 : 0)] : 0.0f;
    a.y = (n < 4) ? W_hh[n * 4 + (half ? 3 : 1)] : 0.0f;

    // per-lane input-bias constants for the two hidden units this lane carries
    const int k0 = 2 * half, k1 = 2 * half + 1;
    const float c0a = b_ih[k0] + b_hh[k0];          // x == 0
    const float c1a = c0a + W_ih[k0];               // x == 1
    const float c0b = b_ih[k1] + b_hh[k1];
    const float c1b = c0b + W_ih[k1];

    float hB0 = 0.0f, hB1 = 0.0f;                   // h0 = zeros
    float latent = 1.0f / (1.0f + expf(-prior[0])); // sigmoid(prior)
    float lossAcc = 0.0f;

    for (int t = 0; t < Tn; ++t) {
        const int idx = t * Bn + chain;
        const float xv = x[idx];
        const float yv = y[idx];                    // issue before tanh chain
        if (t + 1 < Tn) {                           // hide HBM latency of next step
            __builtin_prefetch(&x[idx + Bn], 0, 3);
            __builtin_prefetch(&y[idx + Bn], 0, 3);
        }

        // ---- RNN matvec for 16 chains in one WMMA ----
        v2f bm; bm.x = hB0; bm.y = hB1;
        v8f c = {};
        v8f d = __builtin_amdgcn_wmma_f32_16x16x4_f32(
            /*neg_a=*/false, a, /*neg_b=*/false, bm,
            /*c_mod=*/(short)0, c, /*reuse_a=*/false, /*reuse_b=*/false);

        // re-stripe D rows 0-3 (live in lanes 0-15) back into B layout
        const float s0 = __shfl(d[0], n, 32);
        const float s1 = __shfl(d[1], n, 32);
        const float s2 = __shfl(d[2], n, 32);
        const float s3 = __shfl(d[3], n, 32);
        const float pre0 = (half ? s2 : s0) + (xv != 0.0f ? c1a : c0a);
        const float pre1 = (half ? s3 : s1) + (xv != 0.0f ? c1b : c0b);
        hB0 = fast_tanh(pre0);
        hB1 = fast_tanh(pre1);

        // bring h2/h3 down so lanes 0-15 hold all four params of their chain
        const float h2v = __shfl(hB0, 16 + n, 32);
        const float h3v = __shfl(hB1, 16 + n, 32);

        if (half == 0) {
            // params = (h + 1) / 2 -> l, f, g, s
            const float l = 0.5f * (hB0 + 1.0f);
            const float f = 0.5f * (hB1 + 1.0f);
            const float g = 0.5f * (h2v + 1.0f);
            const float s = 0.5f * (h3v + 1.0f);

            const float num1    = latent * (1.0f - s);
            const float correct = num1 + (1.0f - latent) * g;
            // reference: kt1 = num1/correct; mt = kt1*correct + kt0*(1-correct)
            // kt1*correct == num1, so the first division cancels exactly.
            const float num0    = latent * s;
            const float kt0     = num0 / (num0 + (1.0f - latent) * (1.0f - g));
            const float mt      = num1 + kt0 * (1.0f - correct);
            latent = mt * (1.0f - f) + (1.0f - mt) * l;

            out_correct[idx] = correct;
            out_latent[idx]  = latent;

            // y is exactly 0 or 1: only one of log(p)/log(1-p) contributes
            const float pv = (yv != 0.0f) ? correct : (1.0f - correct);
            lossAcc -= fmaxf(fast_log(pv), -100.0f);
        }
    }

    // ---- deterministic loss reduction: wave -> block partial ----
    for (int off = 16; off > 0; off >>= 1)
        lossAcc += __shfl_down(lossAcc, off, 32);
    __shared__ float red[WAVES_PER_BLK];
    if (lane == 0) red[waveInBlk] = lossAcc;
    __syncthreads();
    if (threadIdx.x == 0) {
        float tot = 0.0f;
        #pragma unroll
        for (int i = 0; i < WAVES_PER_BLK; ++i) tot += red[i];
        block_loss[blockIdx.x] = tot;
    }
}

__global__ void bkt_reduce_loss(const float* __restrict__ block_loss,
                                float* __restrict__ out_loss)
{
    if (threadIdx.x == 0) {
        float tot = 0.0f;
        #pragma unroll
        for (int i = 0; i < NBLOCKS; ++i) tot += block_loss[i];
        *out_loss = tot * (1.0f / ((float)Tn * (float)Bn));
    }
}

extern "C" void kernel_launch(void* const* d_in, const int* in_sizes, int n_in,
                              void* d_out, int out_size, void* d_ws, size_t ws_size,
                              hipStream_t stream) {
    const float* x     = (const float*)d_in[0];   // (T, B, 1)
    const float* y     = (const float*)d_in[1];   // (T, B, 1)
    const float* prior = (const float*)d_in[2];   // (1,)
    const float* W_ih  = (const float*)d_in[3];   // (4, 1)
    const float* W_hh  = (const float*)d_in[4];   // (4, 4) row-major
    const float* b_ih  = (const float*)d_in[5];   // (4,)
    const float* b_hh  = (const float*)d_in[6];   // (4,)

    float* out       = (float*)d_out;
    float* corrects  = out;                                // T*B
    float* latents   = out + (size_t)Tn * Bn;              // T*B
    float* loss      = out + 2 * (size_t)Tn * Bn;          // 1
    float* blockLoss = (float*)d_ws;                       // NBLOCKS floats

    bkt_rnn_fused<<<NBLOCKS, WAVES_PER_BLK * 32, 0, stream>>>(
        x, y, prior, W_ih, W_hh, b_ih, b_hh, corrects, latents, blockLoss);
    bkt_reduce_loss<<<1, 32, 0, stream>>>(blockLoss, loss);
}